// MPNN_vertex_GNN_Power_86878598464000
// MI455X (gfx1250) — compile-verified
//
#include <hip/hip_runtime.h>

typedef __attribute__((ext_vector_type(2))) float v2f;
typedef __attribute__((ext_vector_type(8))) float v8f;

// Problem constants (from setup_inputs)
#define NB    1024      // B = BATCH*nbr = 64*16
#define LL    64
#define KK    32
#define UU    4
#define IN1   64
#define IN2   8
#define OUTC  32
#define XSTR  68        // padded LDS row stride for transposed X (conflict-free b64)
#define EPS_F 1e-5f

// --------------------------- async LDS copy path ---------------------------
#if defined(__HIP_DEVICE_COMPILE__) && __has_builtin(__builtin_amdgcn_global_load_async_to_lds_b32)
#define GNN_ASYNC_LDS 1
#else
#define GNN_ASYNC_LDS 0
#endif

#if GNN_ASYNC_LDS
typedef __attribute__((address_space(1))) int gnn_as1_int;
typedef __attribute__((address_space(3))) int gnn_as3_int;
#endif

__device__ __forceinline__ void gnn_copy_b32(const float* g, float* l) {
#if GNN_ASYNC_LDS
    __builtin_amdgcn_global_load_async_to_lds_b32(
        (gnn_as1_int*)g, (gnn_as3_int*)l, 0, 0);
#else
    *l = *g;
#endif
}

__device__ __forceinline__ void gnn_wait_async() {
#if GNN_ASYNC_LDS
#if __has_builtin(__builtin_amdgcn_s_wait_asynccnt)
    __builtin_amdgcn_s_wait_asynccnt(0);
#else
    asm volatile("s_wait_asynccnt 0x0" ::: "memory");
#endif
#endif
}

// ---------------------------------------------------------------------------
// K1: H means. H[b,c,l,k] = (c<4 ? Hr : Hi)[b,l,k,c&3]
//     hmk[b,c,l] = mean_k H (feeds AP branch)   [B,8,64]
//     hml[b,c,k] = mean_l H (feeds UE branch)   [B,8,32]
// ---------------------------------------------------------------------------
__global__ __launch_bounds__(256) void gnn_hmeans_kernel(
    const float* __restrict__ Hr, const float* __restrict__ Hi,
    float* __restrict__ hmk_g, float* __restrict__ hml_g)
{
    const int b = blockIdx.x;
    const int t = threadIdx.x;
    const float* base_r = Hr + (size_t)b * (LL * KK * UU);
    const float* base_i = Hi + (size_t)b * (LL * KK * UU);

    for (int p = t; p < 8 * LL; p += 256) {
        int c = p >> 6, l = p & 63, u = c & 3;
        const float* src = (c < 4 ? base_r : base_i) + l * (KK * UU) + u;
        float s = 0.f;
        #pragma unroll
        for (int k = 0; k < KK; ++k) s += src[k * UU];
        hmk_g[b * 512 + p] = s * (1.f / KK);
    }
    {
        int c = t >> 5, k = t & 31, u = c & 3;
        const float* src = (c < 4 ? base_r : base_i) + k * UU + u;
        float s = 0.f;
        #pragma unroll
        for (int l = 0; l < LL; ++l) s += src[l * (KK * UU)];
        hml_g[b * 256 + t] = s * (1.f / LL);
    }
}

// ---------------------------------------------------------------------------
// K2: layer 1 (A inputs are zero -> only 0.1*P@Hmean), ReLU,
//     per-block BN stats partials (deterministic two-pass reduction).
//     part per block: [0:64)=ap_sum [64:128)=ap_sq [128:192)=ue_sum [192:256)=ue_sq
// ---------------------------------------------------------------------------
__global__ __launch_bounds__(256) void gnn_layer1_kernel(
    const float* __restrict__ hmk_g, const float* __restrict__ hml_g,
    const float* __restrict__ Ph,
    float* __restrict__ A1ap, float* __restrict__ A1ue,
    float* __restrict__ part)
{
    __shared__ float hmk[8 * LL];
    __shared__ float hml[8 * KK];
    __shared__ float Yap[IN1 * LL];
    __shared__ float Yue[IN1 * KK];

    const int b = blockIdx.x;
    const int t = threadIdx.x;
    const float* P1a = Ph;                 // layer 0, P1_AP [64,8]
    const float* P1u = Ph + IN1 * IN2;     // layer 0, P1_UE [64,8]

    for (int i = t; i < 8 * LL; i += 256) gnn_copy_b32(hmk_g + b * 512 + i, &hmk[i]);
    gnn_copy_b32(hml_g + b * 256 + t, &hml[t]);
    gnn_wait_async();
    __syncthreads();

    for (int i = t; i < IN1 * LL; i += 256) {
        int c = i >> 6, l = i & 63;
        float s = 0.f;
        #pragma unroll
        for (int j = 0; j < IN2; ++j) s += P1a[c * IN2 + j] * hmk[j * LL + l];
        Yap[i] = fmaxf(0.1f * s, 0.f);
    }
    for (int i = t; i < IN1 * KK; i += 256) {
        int c = i >> 5, k = i & 31;
        float s = 0.f;
        #pragma unroll
        for (int j = 0; j < IN2; ++j) s += P1u[c * IN2 + j] * hml[j * KK + k];
        Yue[i] = fmaxf(0.1f * s, 0.f);
    }
    __syncthreads();

    if (t < IN1) {
        float s = 0.f, sq = 0.f;
        for (int l = 0; l < LL; ++l) { float v = Yap[t * LL + l]; s += v; sq += v * v; }
        part[b * 256 + t] = s;
        part[b * 256 + 64 + t] = sq;
        float su = 0.f, squ = 0.f;
        for (int k = 0; k < KK; ++k) { float v = Yue[t * KK + k]; su += v; squ += v * v; }
        part[b * 256 + 128 + t] = su;
        part[b * 256 + 192 + t] = squ;
    }
    for (int i = t; i < IN1 * LL; i += 256) A1ap[b * (IN1 * LL) + i] = Yap[i];
    for (int i = t; i < IN1 * KK; i += 256) A1ue[b * (IN1 * KK) + i] = Yue[i];
}

// ---------------------------------------------------------------------------
// K3/K5: fold global BN stats -> per-channel affine (scale, shift)
//     aff: [0:64)=ap_scale [64:128)=ap_shift [128:192)=ue_scale [192:256)=ue_shift
// ---------------------------------------------------------------------------
__global__ void gnn_bn_finalize_kernel(
    const float* __restrict__ part,
    const float* __restrict__ gamma, const float* __restrict__ beta,
    float* __restrict__ aff, float nap_inv, float nue_inv)
{
    const int t = threadIdx.x;
    if (t >= IN1) return;
    float s = 0.f, sq = 0.f, su = 0.f, squ = 0.f;
    for (int b = 0; b < NB; ++b) {
        const float* p = part + b * 256;
        s += p[t]; sq += p[64 + t]; su += p[128 + t]; squ += p[192 + t];
    }
    float m  = s * nap_inv;
    float v  = sq * nap_inv - m * m;
    float sc = gamma[t] * rsqrtf(v + EPS_F);
    aff[t] = sc;
    aff[64 + t] = beta[t] - m * sc;
    float mu  = su * nue_inv;
    float vu  = squ * nue_inv - mu * mu;
    float scu = gamma[t] * rsqrtf(vu + EPS_F);
    aff[128 + t] = scu;
    aff[192 + t] = beta[t] - mu * scu;
}

// ---------------------------------------------------------------------------
// Fold BN affine into weights: Qf[o,i] = Q[o,i]*scale[i];
// qshift[o] = sum_i Q[o,i]*shift[i]  (so  Q @ (x*s + h) = Qf @ x + qshift)
// ---------------------------------------------------------------------------
__global__ void gnn_fold_kernel(
    const float* __restrict__ Q, const float* __restrict__ aff, int scale_off,
    int rows, float* __restrict__ Qf, float* __restrict__ qshift)
{
    const float* scale = aff + scale_off;
    const float* shift = scale + 64;
    const int t = threadIdx.x;
    for (int i = t; i < rows * IN1; i += 256)
        Qf[i] = Q[i] * scale[i & 63];
    if (t < rows) {
        float s = 0.f;
        for (int i = 0; i < IN1; ++i) s += Q[t * IN1 + i] * shift[i];
        qshift[t] = s;
    }
}

// ---------------------------------------------------------------------------
// WMMA helper: one 16x16 f32 tile of C = Qf[M x 64] * X[64 x N], with X stored
// transposed in LDS as Xt[n][k], row stride XSTR.
//   A 16x4 frag: m = lane%16, k = v + 2*(lane/16)   -> contiguous b64 from Qf
//   B 4x16 frag: n = lane%16, k = v + 2*(lane/16)   -> contiguous ds b64 from Xt
//   C v8f:       m = r + 8*(lane/16), n = lane%16
// ---------------------------------------------------------------------------
__device__ __forceinline__ v8f gnn_wmma_tile(
    const float* __restrict__ Qf, const float* Xt, int ti, int tj,
    int lrow, int lhi)
{
    const float* qrow = Qf + (ti * 16 + lrow) * IN1 + 2 * lhi;
    const float* xrow = Xt + (tj * 16 + lrow) * XSTR + 2 * lhi;
    v8f acc = {};
    #pragma unroll
    for (int kt = 0; kt < 16; ++kt) {
        v2f a   = *(const v2f*)(qrow + kt * 4);
        v2f bfr = *(const v2f*)(xrow + kt * 4);
        acc = __builtin_amdgcn_wmma_f32_16x16x4_f32(
            false, a, false, bfr, (short)0, acc, false, false);
    }
    return acc;
}

// ---------------------------------------------------------------------------
// K4: layer 2 (full): async-copy raw activations to LDS transposed,
//     WMMA GEMMs with BN-folded weights, Q2*mean + 0.1*P*Hmean, ReLU,
//     store + BN stats partials.
// ---------------------------------------------------------------------------
__global__ __launch_bounds__(256) void gnn_layer2_kernel(
    const float* __restrict__ A1ap, const float* __restrict__ A1ue,
    const float* __restrict__ hmk_g, const float* __restrict__ hml_g,
    const float* __restrict__ Qh, const float* __restrict__ Ph,
    const float* __restrict__ aff1,
    const float* __restrict__ Q1af, const float* __restrict__ qsa,
    const float* __restrict__ Q1uf, const float* __restrict__ qsu,
    float* __restrict__ A2ap, float* __restrict__ A2ue,
    float* __restrict__ part)
{
    __shared__ float XapT[LL * XSTR];    // [n=l][k=i] raw A_AP, transposed
    __shared__ float XueT[KK * XSTR];    // [n=k][k=i] raw A_UE, transposed
    __shared__ float Yap[IN1 * LL];
    __shared__ float Yue[IN1 * KK];
    __shared__ float hmk[8 * LL];
    __shared__ float hml[8 * KK];
    __shared__ float abar_ap[IN1], abar_ue[IN1], c2ap[IN1], c2ue[IN1];

    const int b = blockIdx.x;
    const int t = threadIdx.x;

    for (int i = t; i < IN1 * LL; i += 256) {
        int c = i >> 6, l = i & 63;
        gnn_copy_b32(A1ap + b * (IN1 * LL) + i, &XapT[l * XSTR + c]);
    }
    for (int i = t; i < IN1 * KK; i += 256) {
        int c = i >> 5, k = i & 31;
        gnn_copy_b32(A1ue + b * (IN1 * KK) + i, &XueT[k * XSTR + c]);
    }
    for (int i = t; i < 8 * LL; i += 256) gnn_copy_b32(hmk_g + b * 512 + i, &hmk[i]);
    gnn_copy_b32(hml_g + b * 256 + t, &hml[t]);
    gnn_wait_async();
    __syncthreads();

    // BN-affined means over the spatial axis (affine commutes with mean)
    if (t < 64) {
        float s = 0.f;
        for (int l = 0; l < LL; ++l) s += XapT[l * XSTR + t];
        abar_ap[t] = (s * (1.f / LL)) * aff1[t] + aff1[64 + t];
    } else if (t < 128) {
        int c = t - 64; float s = 0.f;
        for (int k = 0; k < KK; ++k) s += XueT[k * XSTR + c];
        abar_ue[c] = (s * (1.f / KK)) * aff1[128 + c] + aff1[192 + c];
    }
    __syncthreads();

    const float* Q2a = Qh + 5 * (IN1 * IN1);   // Qs_hidden[1,1]
    const float* Q2u = Qh + 7 * (IN1 * IN1);   // Qs_hidden[1,3]
    const float* P1a = Ph + 2 * (IN1 * IN2);   // Ps_hidden[1,0]
    const float* P1u = Ph + 3 * (IN1 * IN2);   // Ps_hidden[1,1]

    if (t < 64) {
        float s = 0.f;
        for (int i = 0; i < IN1; ++i) s += Q2a[t * IN1 + i] * abar_ue[i];
        c2ap[t] = s;
    } else if (t < 128) {
        int o = t - 64; float s = 0.f;
        for (int i = 0; i < IN1; ++i) s += Q2u[o * IN1 + i] * abar_ap[i];
        c2ue[o] = s;
    }
    __syncthreads();

    // 24 tiles: 16 AP (4x4 of 64x64) + 8 UE (4x2 of 64x32); 3 per wave
    const int wave = t >> 5, lane = t & 31;
    const int lrow = lane & 15, lhi = lane >> 4;
    for (int tt = wave * 3; tt < wave * 3 + 3; ++tt) {
        const float *Qf, *Xt, *P, *hm, *c2, *qs;
        float* Y;
        int ti, tj, nstride;
        if (tt < 16) {
            ti = tt >> 2; tj = tt & 3; nstride = LL;
            Qf = Q1af; Xt = XapT; Y = Yap; P = P1a; hm = hmk; c2 = c2ap; qs = qsa;
        } else {
            int t2 = tt - 16;
            ti = t2 >> 1; tj = t2 & 1; nstride = KK;
            Qf = Q1uf; Xt = XueT; Y = Yue; P = P1u; hm = hml; c2 = c2ue; qs = qsu;
        }
        v8f acc = gnn_wmma_tile(Qf, Xt, ti, tj, lrow, lhi);
        const int n = tj * 16 + lrow;
        #pragma unroll
        for (int r = 0; r < 8; ++r) {
            int o = ti * 16 + r + 8 * lhi;
            float pt = 0.f;
            #pragma unroll
            for (int j = 0; j < IN2; ++j) pt += P[o * IN2 + j] * hm[j * nstride + n];
            float v = 2.f * (acc[r] + qs[o]) + 2.f * c2[o] + 0.1f * pt;
            Y[o * nstride + n] = fmaxf(v, 0.f);
        }
    }
    __syncthreads();

    if (t < IN1) {
        float s = 0.f, sq = 0.f;
        for (int l = 0; l < LL; ++l) { float v = Yap[t * LL + l]; s += v; sq += v * v; }
        part[b * 256 + t] = s;
        part[b * 256 + 64 + t] = sq;
        float su = 0.f, squ = 0.f;
        for (int k = 0; k < KK; ++k) { float v = Yue[t * KK + k]; su += v; squ += v * v; }
        part[b * 256 + 128 + t] = su;
        part[b * 256 + 192 + t] = squ;
    }
    for (int i = t; i < IN1 * LL; i += 256) A2ap[b * (IN1 * LL) + i] = Yap[i];
    for (int i = t; i < IN1 * KK; i += 256) A2ue[b * (IN1 * KK) + i] = Yue[i];
}

// ---------------------------------------------------------------------------
// K6: output layer (AP branch, no act/BN, BN folded into Qof), per-k L2 norm
//     over l, then fused Fhat = Vhat * Phat write (non-temporal).
// ---------------------------------------------------------------------------
__global__ __launch_bounds__(256) void gnn_output_kernel(
    const float* __restrict__ A2ap, const float* __restrict__ A2ue,
    const float* __restrict__ hmk_g,
    const float* __restrict__ Qo, const float* __restrict__ Po,
    const float* __restrict__ aff2,
    const float* __restrict__ Qof, const float* __restrict__ qso,
    const float* __restrict__ Vhat, float* __restrict__ out)
{
    __shared__ float XapT[LL * XSTR];
    __shared__ float hmk[8 * LL];
    __shared__ float Vsh[OUTC * LL];     // V_AP_out[o=k][l]
    __shared__ float abar_ue[IN1];
    __shared__ float c2[OUTC];
    __shared__ float pscale[OUTC];

    const int b = blockIdx.x;
    const int t = threadIdx.x;

    for (int i = t; i < IN1 * LL; i += 256) {
        int c = i >> 6, l = i & 63;
        gnn_copy_b32(A2ap + b * (IN1 * LL) + i, &XapT[l * XSTR + c]);
    }
    for (int i = t; i < 8 * LL; i += 256) gnn_copy_b32(hmk_g + b * 512 + i, &hmk[i]);
    gnn_wait_async();
    if (t < 64) {
        float s = 0.f;
        for (int k = 0; k < KK; ++k) s += A2ue[b * (IN1 * KK) + t * KK + k];
        abar_ue[t] = (s * (1.f / KK)) * aff2[128 + t] + aff2[192 + t];
    }
    __syncthreads();

    const float* Q2 = Qo + OUTC * IN1;        // Qs_out[1]
    const float* P1 = Po;                     // Ps_out[0]  [32,8]
    if (t < OUTC) {
        float s = 0.f;
        for (int i = 0; i < IN1; ++i) s += Q2[t * IN1 + i] * abar_ue[i];
        c2[t] = s;
    }
    __syncthreads();

    // 8 tiles (2x4 of 32x64), one per wave
    const int wave = t >> 5, lane = t & 31;
    const int lrow = lane & 15, lhi = lane >> 4;
    {
        const int ti = wave >> 2, tj = wave & 3;
        v8f acc = gnn_wmma_tile(Qof, XapT, ti, tj, lrow, lhi);
        const int n = tj * 16 + lrow;
        #pragma unroll
        for (int r = 0; r < 8; ++r) {
            int o = ti * 16 + r + 8 * lhi;
            float pt = 0.f;
            #pragma unroll
            for (int j = 0; j < IN2; ++j) pt += P1[o * IN2 + j] * hmk[j * LL + n];
            Vsh[o * LL + n] = 2.f * (acc[r] + qso[o]) + 2.f * c2[o] + 0.1f * pt;
        }
    }
    __syncthreads();

    if (t < OUTC) {
        float s = 0.f;
        for (int l = 0; l < LL; ++l) { float v = Vsh[t * LL + l]; s += v * v; }
        pscale[t] = 8.0f * rsqrtf(s);        // sqrt(L)/norm, L = 64
    }
    __syncthreads();

    const size_t base = (size_t)b * (LL * KK * UU);
    for (int i = t; i < LL * KK * UU; i += 256) {
        int k = (i >> 2) & 31, l = i >> 7;   // i = l*128 + k*4 + u
        float val = Vhat[base + i] * Vsh[k * LL + l] * pscale[k];
        __builtin_nontemporal_store(val, &out[base + i]);
    }
}

// ---------------------------------------------------------------------------
// Host launcher
// ---------------------------------------------------------------------------
extern "C" void kernel_launch(void* const* d_in, const int* in_sizes, int n_in,
                              void* d_out, int out_size, void* d_ws, size_t ws_size,
                              hipStream_t stream) {
    const float* Hr    = (const float*)d_in[0];
    const float* Hi    = (const float*)d_in[1];
    const float* Vhat  = (const float*)d_in[2];
    const float* Qh    = (const float*)d_in[3];   // [2,4,64,64]
    const float* Ph    = (const float*)d_in[4];   // [2,2,64,8]
    const float* Qo    = (const float*)d_in[5];   // [4,32,64]
    const float* Po    = (const float*)d_in[6];   // [2,32,8]
    const float* gamma = (const float*)d_in[7];   // [2,64]
    const float* beta  = (const float*)d_in[8];   // [2,64]
    float* out = (float*)d_out;

    // Workspace layout (floats)
    float* ws   = (float*)d_ws;
    float* hmk  = ws;                         // 1024*512
    float* hml  = hmk  + NB * 512;            // 1024*256
    float* A1ap = hml  + NB * 256;            // 1024*4096
    float* A1ue = A1ap + NB * IN1 * LL;       // 1024*2048
    float* A2ap = A1ue + NB * IN1 * KK;       // 1024*4096
    float* A2ue = A2ap + NB * IN1 * LL;       // 1024*2048
    float* part = A2ue + NB * IN1 * KK;       // 1024*256 (reused for both BN passes)
    float* aff1 = part + NB * 256;            // 256
    float* aff2 = aff1 + 256;                 // 256
    float* Q1af = aff2 + 256;                 // 4096
    float* Q1uf = Q1af + IN1 * IN1;           // 4096
    float* Qof  = Q1uf + IN1 * IN1;           // 2048
    float* qsa  = Qof  + OUTC * IN1;          // 64
    float* qsu  = qsa  + IN1;                 // 64
    float* qso  = qsu  + IN1;                 // 32

    const float nap_inv = 1.0f / (float)(NB * LL);   // 65536
    const float nue_inv = 1.0f / (float)(NB * KK);   // 32768

    gnn_hmeans_kernel<<<NB, 256, 0, stream>>>(Hr, Hi, hmk, hml);
    gnn_layer1_kernel<<<NB, 256, 0, stream>>>(hmk, hml, Ph, A1ap, A1ue, part);
    gnn_bn_finalize_kernel<<<1, 64, 0, stream>>>(part, gamma, beta, aff1, nap_inv, nue_inv);
    gnn_fold_kernel<<<1, 256, 0, stream>>>(Qh + 4 * IN1 * IN1, aff1, 0,   IN1,  Q1af, qsa);
    gnn_fold_kernel<<<1, 256, 0, stream>>>(Qh + 6 * IN1 * IN1, aff1, 128, IN1,  Q1uf, qsu);
    gnn_layer2_kernel<<<NB, 256, 0, stream>>>(A1ap, A1ue, hmk, hml, Qh, Ph, aff1,
                                              Q1af, qsa, Q1uf, qsu, A2ap, A2ue, part);
    gnn_bn_finalize_kernel<<<1, 64, 0, stream>>>(part, gamma + 64, beta + 64, aff2, nap_inv, nue_inv);
    gnn_fold_kernel<<<1, 256, 0, stream>>>(Qo, aff2, 0, OUTC, Qof, qso);
    gnn_output_kernel<<<NB, 256, 0, stream>>>(A2ap, A2ue, hmk, Qo, Po, aff2,
                                              Qof, qso, Vhat, out);
}